// GCN_28827820491150
// MI455X (gfx1250) — compile-verified
//
#include <hip/hip_runtime.h>

#define NNODES 50000
#define NEDGES 800000
#define D 64
#define NEG_SLOPE 0.01f

typedef __attribute__((ext_vector_type(2))) float v2f;
typedef __attribute__((ext_vector_type(8))) float v8f;

// ---------------- degree / norm kernels ----------------

__global__ void gcn_zero_f32(float* __restrict__ p, int n) {
  int i = blockIdx.x * blockDim.x + threadIdx.x;
  if (i < n) p[i] = 0.0f;
}

__global__ void gcn_deg_kernel(const int* __restrict__ src, const int* __restrict__ dst,
                               float* __restrict__ outdeg, float* __restrict__ indeg, int nE) {
  int e = blockIdx.x * blockDim.x + threadIdx.x;
  if (e >= nE) return;
  int s = src[e], d = dst[e];
  if (s != d) {
    atomicAdd(&outdeg[s], 1.0f);
    atomicAdd(&indeg[d], 1.0f);
  }
}

__global__ void gcn_rsqrt_kernel(float* __restrict__ p, int n) {
  int i = blockIdx.x * blockDim.x + threadIdx.x;
  if (i < n) p[i] = rsqrtf(p[i] + 1.0f);
}

// hs = h * inv_out (per-row), agg initialized to hs (self-loop term)
__global__ void gcn_scale_init_kernel(const float* __restrict__ h,
                                      const float* __restrict__ inv_out,
                                      float* __restrict__ hs, float* __restrict__ agg,
                                      int total) {
  int i = blockIdx.x * blockDim.x + threadIdx.x;
  if (i >= total) return;
  float v = h[i] * inv_out[i >> 6];   // D == 64
  hs[i] = v;
  agg[i] = v;
}

// edge-parallel scatter-add: agg[dst] += hs[src] for src != dst
__global__ void gcn_scatter_kernel(const float* __restrict__ hs,
                                   const int* __restrict__ src, const int* __restrict__ dst,
                                   float* __restrict__ agg, int nE) {
  int t = blockIdx.x * blockDim.x + threadIdx.x;
  int e = t >> 6;        // 64 feature lanes per edge
  int f = t & 63;
  if (e >= nE) return;
  int s = src[e], d = dst[e];
  if (s == d) return;
  atomicAdd(&agg[(size_t)d * D + f], hs[(size_t)s * D + f]);
}

// ---------------- WMMA GEMM: out = LeakyReLU((agg * inv_in) @ W + b) ----------------
// Block: 128 threads = 4 waves; block handles 16 rows; wave w handles cols [16w,16w+16).
// K loop: 16 steps of V_WMMA_F32_16X16X4_F32 (full f32 precision).

__global__ __launch_bounds__(128)
void gcn_gemm_wmma_kernel(const float* __restrict__ agg, const float* __restrict__ inv_in,
                          const float* __restrict__ W, const float* __restrict__ bias,
                          float* __restrict__ out) {
  __shared__ float Wlds[D * D];   // 16 KB of the 320 KB/WGP LDS

  int tid = threadIdx.x;
  // Stage W (64x64 f32, row-major) into LDS with float4 loads.
  {
    const float4* Wv = (const float4*)W;
    float4* Lv = (float4*)Wlds;
#pragma unroll
    for (int i = 0; i < 8; ++i)
      Lv[tid + i * 128] = Wv[tid + i * 128];
  }
  __syncthreads();

  const int wave = tid >> 5;          // 0..3  -> column tile
  const int lane = tid & 31;
  const int half = lane >> 4;         // 0: K pair {0,1}, 1: K pair {2,3}
  const int l16  = lane & 15;
  const int row0 = blockIdx.x * 16;
  const int colBase = wave * 16;

  const int arow = row0 + l16;        // A-fragment row for this lane (M = lane%16)
  const float rscale = inv_in[arow];  // fold dst-side norm into A load
  const float2* ap = (const float2*)(agg + (size_t)arow * D);

  v8f c = {0.f, 0.f, 0.f, 0.f, 0.f, 0.f, 0.f, 0.f};

#pragma unroll
  for (int k = 0; k < 16; ++k) {      // K = 64 in steps of 4
    // A 16x4: lane holds A[M=l16, K = 4k + 2*half + {0,1}]
    float2 av = ap[k * 2 + half];
    v2f a; a.x = av.x * rscale; a.y = av.y * rscale;
    // B 4x16: lane holds W[K = 4k + 2*half + {0,1}, N = colBase + l16]
    int kr = k * 4 + half * 2;
    v2f b;
    b.x = Wlds[kr * D + colBase + l16];
    b.y = Wlds[(kr + 1) * D + colBase + l16];
#if __has_builtin(__builtin_amdgcn_wmma_f32_16x16x4_f32)
    c = __builtin_amdgcn_wmma_f32_16x16x4_f32(false, a, false, b,
                                              (short)0, c, false, false);
#else
    // host-pass / fallback placeholder (never the intended device path)
    c[0] += a.x * b.x + a.y * b.y;
#endif
  }

  const float bv = bias[colBase + l16];
#pragma unroll
  for (int v = 0; v < 8; ++v) {
    // D layout: VGPR v -> M = v + 8*half, N = l16
    float x = c[v] + bv;
    x = (x > 0.0f) ? x : x * NEG_SLOPE;
    out[(size_t)(row0 + half * 8 + v) * D + colBase + l16] = x;
  }
}

// ---------------- launch ----------------

extern "C" void kernel_launch(void* const* d_in, const int* in_sizes, int n_in,
                              void* d_out, int out_size, void* d_ws, size_t ws_size,
                              hipStream_t stream) {
  const float* in_feat = (const float*)d_in[0];
  const float* W1      = (const float*)d_in[1];
  const float* b1      = (const float*)d_in[2];
  const float* W2      = (const float*)d_in[3];
  const float* b2      = (const float*)d_in[4];
  const int*   src     = (const int*)d_in[5];
  const int*   dst     = (const int*)d_in[6];
  float* out = (float*)d_out;

  float* ws      = (float*)d_ws;
  float* inv_out = ws;                      // [N]
  float* inv_in  = ws + NNODES;             // [N]
  float* buf0    = ws + 2 * NNODES;         // [N*D]  hs
  float* buf1    = buf0 + (size_t)NNODES * D; // [N*D] agg / h1

  const int NT = 256;
  // degrees -> inv-sqrt norms (inv_out, inv_in contiguous: one rsqrt pass)
  gcn_zero_f32<<<(2 * NNODES + NT - 1) / NT, NT, 0, stream>>>(inv_out, 2 * NNODES);
  gcn_deg_kernel<<<(NEDGES + NT - 1) / NT, NT, 0, stream>>>(src, dst, inv_out, inv_in, NEDGES);
  gcn_rsqrt_kernel<<<(2 * NNODES + NT - 1) / NT, NT, 0, stream>>>(inv_out, 2 * NNODES);

  const int totalF = NNODES * D;
  const int scatterThreads = NEDGES * D;

  // ---- layer 1: hs->buf0, agg->buf1, gemm in-place buf1 ----
  gcn_scale_init_kernel<<<(totalF + NT - 1) / NT, NT, 0, stream>>>(in_feat, inv_out, buf0, buf1, totalF);
  gcn_scatter_kernel<<<(scatterThreads + NT - 1) / NT, NT, 0, stream>>>(buf0, src, dst, buf1, NEDGES);
  gcn_gemm_wmma_kernel<<<NNODES / 16, 128, 0, stream>>>(buf1, inv_in, W1, b1, buf1);

  // ---- layer 2: hs->buf0, agg->d_out, gemm in-place d_out ----
  gcn_scale_init_kernel<<<(totalF + NT - 1) / NT, NT, 0, stream>>>(buf1, inv_out, buf0, out, totalF);
  gcn_scatter_kernel<<<(scatterThreads + NT - 1) / NT, NT, 0, stream>>>(buf0, src, dst, out, NEDGES);
  gcn_gemm_wmma_kernel<<<NNODES / 16, 128, 0, stream>>>(out, inv_in, W2, b2, out);
}